// GNNGraphProcessor_75024488726734
// MI455X (gfx1250) — compile-verified
//
#include <hip/hip_runtime.h>
#include <hip/hip_bf16.h>
#include <math.h>

typedef __attribute__((ext_vector_type(2))) float v2f;
typedef __attribute__((ext_vector_type(8))) float v8f;
typedef __attribute__((ext_vector_type(4))) unsigned int u32x4;
typedef __attribute__((ext_vector_type(8))) int i32x8;
typedef __attribute__((ext_vector_type(4))) int i32x4;

#define N_NODES 50000
#define N_EDGES 800000

// ---------------------------------------------------------------------------
// fill
// ---------------------------------------------------------------------------
__global__ void fill_kernel(float* __restrict__ p, float v, long long n) {
    long long i = (long long)blockIdx.x * blockDim.x + threadIdx.x;
    long long stride = (long long)gridDim.x * blockDim.x;
    for (; i < n; i += stride) p[i] = v;
}

// ---------------------------------------------------------------------------
// degree accumulation: deg[dst] += 1 (deg pre-filled with 1.0 for self-loops)
// ---------------------------------------------------------------------------
__global__ void deg_kernel(const long long* __restrict__ dst, float* __restrict__ deg, int E) {
    int i = blockIdx.x * blockDim.x + threadIdx.x;
    int stride = gridDim.x * blockDim.x;
    for (; i < E; i += stride) {
        atomicAdd(&deg[(int)dst[i]], 1.0f);
    }
}

// dinv[i] = rsqrt(deg[i])   (deg >= 1 always due to self loops)
__global__ void rsqrt_kernel(float* __restrict__ d, int n) {
    int i = blockIdx.x * blockDim.x + threadIdx.x;
    if (i < n) d[i] = rsqrtf(d[i]);
}

// ---------------------------------------------------------------------------
// fp32 GEMM via V_WMMA_F32_16X16X4_F32, compile-time N/K so the 32-step MAC
// loop fully unrolls with immediate-offset loads.
// C[M,N] = A[M,K] * B[K,N], row-major, M%16==0.
// One 256-thread block = 8 waves; block computes a 16-row slab of C;
// wave w computes the 16x16 tile at n0 = (blockIdx.y*8 + w)*16.
// A tile staged in LDS via the Tensor Data Mover with hardware row padding
// (128-DWORD rows + 4-DWORD pad -> stride 132 floats, bank-conflict-free).
// ---------------------------------------------------------------------------
template <int N, int K>
__global__ __launch_bounds__(256) void gemm_wmma_kernel(
    const float* __restrict__ A, const float* __restrict__ B, float* __restrict__ C,
    int M) {
    __shared__ float As[16 * 132];

    const int m0 = blockIdx.x * 16;
    const int tid = threadIdx.x;

#if defined(__gfx1250__) && __has_builtin(__builtin_amdgcn_tensor_load_to_lds)
    // --- TDM: DMA the 16xK fp32 A tile into LDS, padding 4 DWORDs per 128 ---
    if (tid < 32) {  // one wave issues the tensor op (EXEC ignored by TDM)
        const unsigned lds_off = (unsigned)(uintptr_t)&As[0];
        const unsigned long long ga =
            (unsigned long long)(uintptr_t)(A + (unsigned long long)m0 * K);
        u32x4 g0;
        g0[0] = 1u;                                           // count=1 (valid D#)
        g0[1] = lds_off;                                      // lds_addr (bytes)
        g0[2] = (unsigned)ga;                                 // global_addr[31:0]
        g0[3] = (unsigned)((ga >> 32) & 0x1FFFFFFu) | (2u << 30);  // ga[56:32] | type=2
        i32x8 g1;
        g1[0] = (int)((2u << 16)      // data_size = 4 bytes
                    | (1u << 20)      // pad_enable
                    | (6u << 22)      // pad_interval: 128 DWORDs
                    | (3u << 25));    // pad_amount:   4 DWORDs
        g1[1] = (int)((unsigned)K << 16);                       // tensor_dim0[15:0]
        g1[2] = (int)(((unsigned)K >> 16) | ((unsigned)M << 16)); // td0[31:16] | td1[15:0]
        g1[3] = (int)((((unsigned)M >> 16) & 0xFFFFu) | ((unsigned)K << 16)); // td1 hi | tile_dim0=K
        g1[4] = 16;                                             // tile_dim1=16, tile_dim2=0
        g1[5] = K;                                              // tensor_dim0_stride[31:0]
        g1[6] = 0;
        g1[7] = 0;
        const i32x4 gz4 = {0, 0, 0, 0};
        const i32x8 gz8 = {0, 0, 0, 0, 0, 0, 0, 0};
        __builtin_amdgcn_tensor_load_to_lds(g0, g1, gz4, gz4, gz8, 0);
        __builtin_amdgcn_s_wait_tensorcnt(0);
    }
    __syncthreads();
#else
    // fallback: cooperative load of the 16 x K A tile
    for (int idx = tid; idx < 16 * K; idx += 256) {
        int r = idx / K;
        int c = idx - r * K;
        As[r * 132 + c] = A[(long long)(m0 + r) * K + c];
    }
    __syncthreads();
#endif

    const int wave = tid >> 5;
    const int lane = tid & 31;
    const int n0 = (blockIdx.y * 8 + wave) * 16;

    const int lm   = lane & 15;        // M index (A) / N index (B)
    const int kOff = (lane >> 4) * 2;  // lanes 16..31 carry K+2, K+3

    // warm the wave's B panel in cache (out of the MAC loop: no branches there)
    const float* Bp = B + (long long)kOff * N + n0 + lm;
    __builtin_prefetch(Bp,                 0, 1);
    __builtin_prefetch(Bp + (K / 4) * N,   0, 1);
    __builtin_prefetch(Bp + (K / 2) * N,   0, 1);
    __builtin_prefetch(Bp + (3 * K / 4) * N, 0, 1);

    v8f acc = {};
    #pragma unroll
    for (int k = 0; k < K; k += 4) {
        v2f a, b;
        a.x = As[lm * 132 + k + kOff];
        a.y = As[lm * 132 + k + kOff + 1];
        b.x = B[(k + kOff)     * N + n0 + lm];
        b.y = B[(k + kOff + 1) * N + n0 + lm];
        acc = __builtin_amdgcn_wmma_f32_16x16x4_f32(
            /*neg_a=*/false, a, /*neg_b=*/false, b,
            /*c_mod=*/(short)0, acc, /*reuse_a=*/false, /*reuse_b=*/false);
    }

    // D layout: VGPR r -> lanes 0..15: M=r, lanes 16..31: M=8+r; N = lane&15
    const int rowHalf = (lane >> 4) * 8;
    const int nIdx = n0 + lm;
    #pragma unroll
    for (int r = 0; r < 8; ++r) {
        C[(long long)(m0 + rowHalf + r) * N + nIdx] = acc[r];
    }
}

// ---------------------------------------------------------------------------
// edge scatter: agg[dst,f] += h[src,f] * dinv[src]*dinv[dst]
// one thread handles one (edge, 4-feature) chunk; f-major => coalesced
// ---------------------------------------------------------------------------
__global__ void scatter_kernel(const float* __restrict__ h,
                               const long long* __restrict__ src,
                               const long long* __restrict__ dst,
                               const float* __restrict__ dinv,
                               float* __restrict__ agg,
                               int E, int F, int fshift /* log2(F/4) */) {
    long long tid = (long long)blockIdx.x * blockDim.x + threadIdx.x;
    const long long total = (long long)E << fshift;
    const long long stride = (long long)gridDim.x * blockDim.x;
    const int fmask = (1 << fshift) - 1;
    for (; tid < total; tid += stride) {
        const int e = (int)(tid >> fshift);
        const int f = ((int)tid & fmask) << 2;
        const int s = (int)src[e];
        const int d = (int)dst[e];
        const float norm = dinv[s] * dinv[d];
        const float4 hv = *reinterpret_cast<const float4*>(h + (long long)s * F + f);
        float* dp = agg + (long long)d * F + f;
        atomicAdd(dp + 0, hv.x * norm);
        atomicAdd(dp + 1, hv.y * norm);
        atomicAdd(dp + 2, hv.z * norm);
        atomicAdd(dp + 3, hv.w * norm);
    }
}

// ---------------------------------------------------------------------------
// finalize: out = relu(agg + h_self * dinv^2 + bias)   (self-loop folded in)
// ---------------------------------------------------------------------------
__global__ void finalize_kernel(const float* __restrict__ agg,
                                const float* __restrict__ hself,
                                const float* __restrict__ dinv,
                                const float* __restrict__ bias,
                                float* __restrict__ out,
                                int Nn, int F, int fsh /* log2(F) */) {
    long long idx = (long long)blockIdx.x * blockDim.x + threadIdx.x;
    const long long total = (long long)Nn << fsh;
    const long long stride = (long long)gridDim.x * blockDim.x;
    for (; idx < total; idx += stride) {
        const int i = (int)(idx >> fsh);
        const int f = (int)idx & (F - 1);
        const float di = dinv[i];
        const float v = agg[idx] + hself[idx] * di * di + bias[f];
        out[idx] = fmaxf(v, 0.0f);
    }
}

// ---------------------------------------------------------------------------
// logits[i] = dot(emb[i,:256], Wa) + ba   -- one wave32 per node
// ---------------------------------------------------------------------------
__global__ void logits_kernel(const float* __restrict__ emb,
                              const float* __restrict__ Wa,
                              const float* __restrict__ ba,
                              float* __restrict__ logits, int Nn) {
    const int gwave = (blockIdx.x * blockDim.x + threadIdx.x) >> 5;
    const int lane = threadIdx.x & 31;
    if (gwave >= Nn) return;
    const float* row = emb + (long long)gwave * 256;
    float s = 0.0f;
    #pragma unroll
    for (int f = lane; f < 256; f += 32) s += row[f] * Wa[f];
    #pragma unroll
    for (int off = 16; off > 0; off >>= 1) s += __shfl_down(s, off, 32);
    if (lane == 0) logits[gwave] = s + ba[0];
}

// single-block max reduction -> scal[0]
__global__ __launch_bounds__(1024) void max_kernel(const float* __restrict__ logits,
                                                   int Nn, float* __restrict__ scal) {
    __shared__ float red[1024];
    float m = -INFINITY;
    for (int i = threadIdx.x; i < Nn; i += 1024) m = fmaxf(m, logits[i]);
    red[threadIdx.x] = m;
    __syncthreads();
    for (int s = 512; s > 0; s >>= 1) {
        if (threadIdx.x < s) red[threadIdx.x] = fmaxf(red[threadIdx.x], red[threadIdx.x + s]);
        __syncthreads();
    }
    if (threadIdx.x == 0) scal[0] = red[0];
}

// single-block: w[i] = exp(logit[i]-max), scal[1] = sum(w)
__global__ __launch_bounds__(1024) void expsum_kernel(const float* __restrict__ logits,
                                                      float* __restrict__ w,
                                                      int Nn, float* __restrict__ scal) {
    __shared__ float red[1024];
    const float mx = scal[0];
    float s = 0.0f;
    for (int i = threadIdx.x; i < Nn; i += 1024) {
        const float e = __expf(logits[i] - mx);
        w[i] = e;
        s += e;
    }
    red[threadIdx.x] = s;
    __syncthreads();
    for (int t = 512; t > 0; t >>= 1) {
        if (threadIdx.x < t) red[threadIdx.x] += red[threadIdx.x + t];
        __syncthreads();
    }
    if (threadIdx.x == 0) scal[1] = red[0];
}

// gout[f] += sum_i emb[i,f]*w[i] / sum   (gout pre-zeroed)
__global__ __launch_bounds__(256) void pool_kernel(const float* __restrict__ emb,
                                                   const float* __restrict__ w,
                                                   const float* __restrict__ scal,
                                                   float* __restrict__ gout, int Nn) {
    const int f = threadIdx.x;  // 256 features
    const float invSum = 1.0f / scal[1];
    float acc = 0.0f;
    for (int i = blockIdx.x; i < Nn; i += gridDim.x) {
        acc += emb[(long long)i * 256 + f] * w[i];
    }
    atomicAdd(&gout[f], acc * invSum);
}

// ---------------------------------------------------------------------------
// launch
// ---------------------------------------------------------------------------
extern "C" void kernel_launch(void* const* d_in, const int* in_sizes, int n_in,
                              void* d_out, int out_size, void* d_ws, size_t ws_size,
                              hipStream_t stream) {
    const float*     x  = (const float*)d_in[0];
    const long long* ei = (const long long*)d_in[1];  // jnp.int64
    const float*     W1 = (const float*)d_in[2];
    const float*     b1 = (const float*)d_in[3];
    const float*     W2 = (const float*)d_in[4];
    const float*     b2 = (const float*)d_in[5];
    const float*     Wa = (const float*)d_in[6];
    const float*     ba = (const float*)d_in[7];

    const long long* src = ei;            // edge_index[0]
    const long long* dst = ei + N_EDGES;  // edge_index[1]

    float* node_out  = (float*)d_out;                              // (N, 256)
    float* graph_out = node_out + (long long)N_NODES * 256;        // (256,)

    // workspace layout (floats)
    float* ws     = (float*)d_ws;
    float* dinv   = ws;                                  // N
    float* h0     = dinv + 50048;                        // N*128  (x@W1)
    float* h1     = h0 + (long long)N_NODES * 128;       // N*128  (agg1 -> relu'd in place)
    float* h2     = h1 + (long long)N_NODES * 128;       // N*256  (h1@W2)
    float* agg2   = h2 + (long long)N_NODES * 256;       // N*256
    float* logits = agg2 + (long long)N_NODES * 256;     // N
    float* wexp   = logits + 50048;                      // N
    float* scal   = wexp + 50048;                        // [max, sum]

    const int THREADS = 256;

    // 1) degrees (init 1.0 = self loop) then rsqrt
    fill_kernel<<<256, THREADS, 0, stream>>>(dinv, 1.0f, (long long)N_NODES);
    deg_kernel<<<(N_EDGES + THREADS - 1) / THREADS, THREADS, 0, stream>>>(dst, dinv, N_EDGES);
    rsqrt_kernel<<<(N_NODES + THREADS - 1) / THREADS, THREADS, 0, stream>>>(dinv, N_NODES);

    // 2) h0 = x @ W1   (50000x128 @ 128x128)
    {
        dim3 grid(N_NODES / 16, 1);
        gemm_wmma_kernel<128, 128><<<grid, THREADS, 0, stream>>>(x, W1, h0, N_NODES);
    }

    // 3) layer-1 aggregation + finalize (relu(agg + self*dinv^2 + b1)) into h1
    fill_kernel<<<2048, THREADS, 0, stream>>>(h1, 0.0f, (long long)N_NODES * 128);
    scatter_kernel<<<8192, THREADS, 0, stream>>>(h0, src, dst, dinv, h1, N_EDGES, 128, 5);
    finalize_kernel<<<4096, THREADS, 0, stream>>>(h1, h0, dinv, b1, h1, N_NODES, 128, 7);

    // 4) h2 = h1 @ W2   (50000x128 @ 128x256)
    {
        dim3 grid(N_NODES / 16, 2);
        gemm_wmma_kernel<256, 128><<<grid, THREADS, 0, stream>>>(h1, W2, h2, N_NODES);
    }

    // 5) layer-2 aggregation + finalize into node_out (d_out)
    fill_kernel<<<4096, THREADS, 0, stream>>>(agg2, 0.0f, (long long)N_NODES * 256);
    scatter_kernel<<<8192, THREADS, 0, stream>>>(h2, src, dst, dinv, agg2, N_EDGES, 256, 6);
    finalize_kernel<<<8192, THREADS, 0, stream>>>(agg2, h2, dinv, b2, node_out, N_NODES, 256, 8);

    // 6) attention pooling
    logits_kernel<<<(N_NODES * 32 + THREADS - 1) / THREADS, THREADS, 0, stream>>>(
        node_out, Wa, ba, logits, N_NODES);
    max_kernel<<<1, 1024, 0, stream>>>(logits, N_NODES, scal);
    expsum_kernel<<<1, 1024, 0, stream>>>(logits, wexp, N_NODES, scal);
    fill_kernel<<<1, 256, 0, stream>>>(graph_out, 0.0f, 256);
    pool_kernel<<<256, 256, 0, stream>>>(node_out, wexp, scal, graph_out, N_NODES);
}